// Quanv2d_86457691668487
// MI455X (gfx1250) — compile-verified
//
#include <hip/hip_runtime.h>
#include <hip/hip_bf16.h>

typedef __attribute__((ext_vector_type(2))) float v2f;
typedef __attribute__((ext_vector_type(8))) float v8f;

#define N_Q 4
#define P_PER_B 3969       // 63*63 patches per image
#define H 128
#define W 128
#define CHW (3*128*128)

// ---------------------------------------------------------------------------
// Build the fixed 16x16 complex unitary U (post-RY part of the circuit) from
// weights[3..6].  U[o][i] = RXprod[o][i] * G[i], where
//   G[i]       = exp(i * sum_q (th_q/2) * (b_q(i)^b_{q+1}(i) ? +1 : -1))
//   RXprod     = (-i)^popcount(o^i) * prod_q (o_q==i_q ? cos(th_q/2) : sin(th_q/2))
// Qubit q maps to bit (3-q) of the 4-bit basis index (state tensor C-order).
// Output: ws[0..255] = Ure row-major, ws[256..511] = Uim row-major.
// ---------------------------------------------------------------------------
__global__ void __launch_bounds__(256) build_u_kernel(const float* __restrict__ w,
                                                      float* __restrict__ ws) {
    const int t = threadIdx.x;
    const int o = t >> 4;
    const int i = t & 15;

    float th[N_Q];
#pragma unroll
    for (int q = 0; q < N_Q; ++q) th[q] = w[q + N_Q - 1];  // indices 3..6

    // diagonal phase from CNOT-RZ-CNOT blocks (column i)
    float alpha = 0.0f;
#pragma unroll
    for (int q = 0; q < N_Q; ++q) {
        const int bq = (i >> (3 - q)) & 1;
        const int bc = (i >> (3 - ((q + 1) & 3))) & 1;
        alpha += 0.5f * th[q] * ((bq ^ bc) ? 1.0f : -1.0f);
    }
    const float ga = __cosf(alpha);
    const float gb = __sinf(alpha);

    // RX tensor-product entry
    float mag = 1.0f;
    int d = 0;
#pragma unroll
    for (int q = 0; q < N_Q; ++q) {
        const int ob = (o >> (3 - q)) & 1;
        const int ib = (i >> (3 - q)) & 1;
        const float c = __cosf(0.5f * th[q]);
        const float s = __sinf(0.5f * th[q]);
        if (ob == ib) mag *= c; else { mag *= s; ++d; }
    }
    // (-i)^d
    float pr, pim;
    switch (d & 3) {
        case 0: pr = 1.0f;  pim = 0.0f;  break;
        case 1: pr = 0.0f;  pim = -1.0f; break;
        case 2: pr = -1.0f; pim = 0.0f;  break;
        default: pr = 0.0f; pim = 1.0f;  break;
    }
    ws[o * 16 + i]       = mag * (pr * ga - pim * gb);
    ws[256 + o * 16 + i] = mag * (pr * gb + pim * ga);
}

// ---------------------------------------------------------------------------
// Main kernel: one wave32 processes 16 patches as one 16x16 WMMA C/D tile.
//   A (16 patches x 16 state)  built from per-patch product states s0
//   B (16 x 16)                = Ure^T / Uim^T   (two chained-WMMA products)
//   probs = Re^2 + Im^2  ->  LDS  ->  +-1 Z-contraction  ->  global store
// Block = 256 threads = 8 waves = 128 patches; grid = totalPatches/128.
// ---------------------------------------------------------------------------
__global__ void __launch_bounds__(256) quanv_wmma_kernel(
        const float* __restrict__ X,
        const float* __restrict__ Ure,
        const float* __restrict__ Uim,
        float* __restrict__ out) {
    const int lane = threadIdx.x & 31;
    const int wave = threadIdx.x >> 5;
    const bool hi  = lane >= 16;
    const int  m   = lane & 15;                       // patch row / B column

    const int waveBase = (blockIdx.x * 8 + wave) * 16;
    const int gp = waveBase + m;                      // this lane's global patch

    // decompose: gp = b*3969 + r*63 + c
    const int b  = gp / P_PER_B;
    const int p  = gp - b * P_PER_B;
    const int r  = p / 63;
    const int cc = p - r * 63;
    const float* xp = X + (size_t)b * CHW + (size_t)(2 * r) * W + (size_t)(2 * cc);

    // product state s0[16] from RY(pi*x) rotations: f_q = (cos,sin)(pi*x_q/2)
    float c0, s0q, c1, s1q, c2, s2q, c3, s3q;
    sincospif(0.5f * xp[0], &s0q, &c0);
    sincospif(0.5f * xp[1], &s1q, &c1);
    sincospif(0.5f * xp[2], &s2q, &c2);
    sincospif(0.5f * xp[3], &s3q, &c3);

    float s0[16];
#pragma unroll
    for (int k = 0; k < 16; ++k) {
        float v = ((k >> 3) & 1) ? s0q : c0;          // qubit 0 = MSB
        v *= ((k >> 2) & 1) ? s1q : c1;
        v *= ((k >> 1) & 1) ? s2q : c2;
        v *= (k & 1) ? s3q : c3;
        s0[k] = v;
    }

    // A tiles: ISA f32 16x4 layout — lanes 0-15: K=0(V0),K=1(V1); lanes 16-31: K=2,K=3
    const int koff = hi ? 2 : 0;
    v2f a[4], bre[4], bim[4];
#pragma unroll
    for (int t = 0; t < 4; ++t) {
        const int k = 4 * t + koff;
        a[t].x = s0[k];
        a[t].y = s0[k + 1];
        // B[k][n] = U[n][k]  (transpose), n = m
        bre[t].x = Ure[m * 16 + k];
        bre[t].y = Ure[m * 16 + k + 1];
        bim[t].x = Uim[m * 16 + k];
        bim[t].y = Uim[m * 16 + k + 1];
    }

    v8f cre = {};
    v8f cim = {};
#pragma unroll
    for (int t = 0; t < 4; ++t) {
        cre = __builtin_amdgcn_wmma_f32_16x16x4_f32(false, a[t], false, bre[t],
                                                    (short)0, cre, false, false);
        cim = __builtin_amdgcn_wmma_f32_16x16x4_f32(false, a[t], false, bim[t],
                                                    (short)0, cim, false, false);
    }

    // probs in C/D layout: VGPR v -> patch row (v or v+8), N = state index = lane%16
    v8f prob;
#pragma unroll
    for (int v = 0; v < 8; ++v)
        prob[v] = cre[v] * cre[v] + cim[v] * cim[v];

    __shared__ float lds[8][16][16];                  // [wave][patch][state]
#pragma unroll
    for (int v = 0; v < 8; ++v) {
        const int mm = v + (hi ? 8 : 0);
        lds[wave][mm][m] = prob[v];
    }
    asm volatile("s_wait_dscnt 0" ::: "memory");      // wave-local DS ordering

    // Z-contraction: z_q = sum_o (1 - 2*b_q(o)) * probs[o]
    // lane handles patch m, qubits (qb, qb+1); qb = 0 for lanes 0-15, 2 for 16-31
    const int qb = hi ? 2 : 0;
    float z0 = 0.0f, z1 = 0.0f;
#pragma unroll
    for (int o = 0; o < 16; ++o) {
        const float pv = lds[wave][m][o];
        z0 += ((o >> (3 - qb)) & 1) ? -pv : pv;
        z1 += ((o >> (2 - qb)) & 1) ? -pv : pv;       // bit (3-(qb+1))
    }

    // out[b, p*4 + q] : flat index = gp*4 + q ; (gp*4+qb) is 8B aligned
    float2 zz = make_float2(z0, z1);
    *reinterpret_cast<float2*>(out + (size_t)gp * 4 + qb) = zz;
}

extern "C" void kernel_launch(void* const* d_in, const int* in_sizes, int n_in,
                              void* d_out, int out_size, void* d_ws, size_t ws_size,
                              hipStream_t stream) {
    const float* X = (const float*)d_in[0];
    const float* w = (const float*)d_in[1];
    float* out = (float*)d_out;
    float* U   = (float*)d_ws;                        // 512 floats used

    const int B = in_sizes[0] / CHW;                  // 128
    const int totalPatches = B * P_PER_B;             // 508032
    const int blocks = totalPatches / 128;            // 3969 (exact)

    hipLaunchKernelGGL(build_u_kernel, dim3(1), dim3(256), 0, stream, w, U);
    hipLaunchKernelGGL(quanv_wmma_kernel, dim3(blocks), dim3(256), 0, stream,
                       X, U, U + 256, out);
}